// CustomLoss_47931835023913
// MI455X (gfx1250) — compile-verified
//
#include <hip/hip_runtime.h>
#include <math.h>
#include <stdint.h>

// ---------------------------------------------------------------------------
// CrossEntropy mean-NLL + conditional correction, MI455X (gfx1250).
// Memory-bound streaming reduction: ~185 MB read @ 23.3 TB/s => ~8us floor.
// CDNA5 async global->LDS copies (ASYNCcnt) with double buffering; NT temporal
// hint since the logit stream is single-use (no reuse => don't pollute caches).
// ---------------------------------------------------------------------------

#define NCLS      10
#define TPB       256
#define GRID1     2048
#define ROWS_PER_THREAD 2                    // one 80B row-pair per lane per tile
#define TILE_ROWS (TPB * ROWS_PER_THREAD)    // 512 rows
#define TILE_BYTES (TILE_ROWS * NCLS * 4)    // 20480 bytes
#define TILE_FLOATS (TILE_BYTES / 4)

// Per-lane async tile load: lane covers bytes t*16 + k*4096, k=0..4.
// INST_OFFSET shifts both LDS and global addresses (ISA 08 pseudocode),
// so LDS is a byte-exact mirror of the 20KB global tile.
__device__ __forceinline__ void async_tile_load(unsigned ldsAddr, unsigned vOff,
                                                unsigned long long sBase) {
    asm volatile("global_load_async_to_lds_b128 %0, %1, %2 offset:0 th:TH_LOAD_NT"
                 :: "v"(ldsAddr), "v"(vOff), "s"(sBase) : "memory");
    asm volatile("global_load_async_to_lds_b128 %0, %1, %2 offset:4096 th:TH_LOAD_NT"
                 :: "v"(ldsAddr), "v"(vOff), "s"(sBase) : "memory");
    asm volatile("global_load_async_to_lds_b128 %0, %1, %2 offset:8192 th:TH_LOAD_NT"
                 :: "v"(ldsAddr), "v"(vOff), "s"(sBase) : "memory");
    asm volatile("global_load_async_to_lds_b128 %0, %1, %2 offset:12288 th:TH_LOAD_NT"
                 :: "v"(ldsAddr), "v"(vOff), "s"(sBase) : "memory");
    asm volatile("global_load_async_to_lds_b128 %0, %1, %2 offset:16384 th:TH_LOAD_NT"
                 :: "v"(ldsAddr), "v"(vOff), "s"(sBase) : "memory");
}

__device__ __forceinline__ void row_nll(const float* __restrict__ x, int tgt,
                                        float& acc, int& flg) {
    float m = x[0];
    int am = 0;
#pragma unroll
    for (int c = 1; c < NCLS; ++c) {
        float v = x[c];
        if (v > m) { m = v; am = c; }     // strict '>' keeps first max (jnp.argmax)
    }
    float s = 0.0f;
#pragma unroll
    for (int c = 0; c < NCLS; ++c) s += __expf(x[c] - m);
    acc += __logf(s) + m - x[tgt];        // -log_softmax(x)[tgt]
    flg |= (int)((tgt == 2) & (am == 3));
}

__global__ __launch_bounds__(TPB) void ce_partial_kernel(
    const float* __restrict__ logits,
    const int*   __restrict__ target,
    float* __restrict__ partial,
    int*   __restrict__ flags,
    int numTiles) {

    __shared__ __align__(16) float tileBuf[2][TILE_FLOATS];   // 2 x 20KB
    __shared__ float rsum[TPB];
    __shared__ int   rflag[TPB];

    const int t = threadIdx.x;
    const unsigned long long base = (unsigned long long)logits;
    // Low 32 bits of a generic pointer to __shared__ == LDS byte offset.
    const unsigned lds0 = (unsigned)(uintptr_t)&tileBuf[0][0] + (unsigned)t * 16u;
    const unsigned lds1 = (unsigned)(uintptr_t)&tileBuf[1][0] + (unsigned)t * 16u;

    float acc = 0.0f;
    int   flg = 0;

    int tile = (int)blockIdx.x;
    int buf = 0;
    if (tile < numTiles) {
        unsigned va = (unsigned)tile * (unsigned)TILE_BYTES + (unsigned)t * 16u;
        async_tile_load(lds0, va, base);
    }

    for (; tile < numTiles; tile += (int)gridDim.x) {
        const int nextTile = tile + (int)gridDim.x;
        if (nextTile < numTiles) {
            // Prefetch next tile into the other buffer, then wait until only
            // those 5 new loads are outstanding (loads complete in order).
            unsigned va = (unsigned)nextTile * (unsigned)TILE_BYTES + (unsigned)t * 16u;
            async_tile_load(buf ? lds0 : lds1, va, base);
            asm volatile("s_wait_asynccnt 5" ::: "memory");
        } else {
            asm volatile("s_wait_asynccnt 0" ::: "memory");
        }
        __syncthreads();   // all waves' async writes for current tile visible

        const float* r = &tileBuf[buf][(unsigned)t * (NCLS * ROWS_PER_THREAD)];
        const int row0 = (tile * TILE_ROWS) + t * ROWS_PER_THREAD;
        row_nll(r,        target[row0],     acc, flg);
        row_nll(r + NCLS, target[row0 + 1], acc, flg);

        __syncthreads();   // done reading this buffer before it is re-filled
        buf ^= 1;
    }

    // Block tree reduction -> one (partial, flag) per block. Deterministic.
    rsum[t]  = acc;
    rflag[t] = flg;
    __syncthreads();
#pragma unroll
    for (int s = TPB / 2; s > 0; s >>= 1) {
        if (t < s) { rsum[t] += rsum[t + s]; rflag[t] |= rflag[t + s]; }
        __syncthreads();
    }
    if (t == 0) {
        partial[blockIdx.x] = rsum[0];
        flags[blockIdx.x]   = rflag[0];
    }
}

__global__ __launch_bounds__(TPB) void ce_final_kernel(
    const float* __restrict__ partial,
    const int*   __restrict__ flags,
    const float* __restrict__ logits,
    const int*   __restrict__ target,
    const int*   __restrict__ epoch_p,
    float* __restrict__ out,
    int nPartials, int tailStart, int nRows) {

    __shared__ float rs[TPB];
    __shared__ int   rf[TPB];
    const int t = threadIdx.x;

    float s = 0.0f;
    int   f = 0;
    for (int i = t; i < nPartials; i += TPB) { s += partial[i]; f |= flags[i]; }

    // Tail rows (none when nRows divides tiles exactly, as here).
    for (int row = tailStart + t; row < nRows; row += TPB) {
        float x[NCLS];
        const float* g = logits + (long long)row * NCLS;
#pragma unroll
        for (int c = 0; c < NCLS; ++c) x[c] = g[c];
        row_nll(x, target[row], s, f);
    }

    rs[t] = s;
    rf[t] = f;
    __syncthreads();
#pragma unroll
    for (int k = TPB / 2; k > 0; k >>= 1) {
        if (t < k) { rs[t] += rs[t + k]; rf[t] |= rf[t + k]; }
        __syncthreads();
    }
    if (t == 0) {
        const float e = (float)epoch_p[0];
        // epoch^(-0.65) / 4^(-3) + 0.01
        const float corr = __powf(e, -0.65f) * 64.0f + 0.01f;
        out[0] = rs[0] / (float)nRows + (rf[0] ? corr : 0.0f);
    }
}

extern "C" void kernel_launch(void* const* d_in, const int* in_sizes, int n_in,
                              void* d_out, int out_size, void* d_ws, size_t ws_size,
                              hipStream_t stream) {
    const float* logits = (const float*)d_in[0];
    const int*   target = (const int*)d_in[1];
    const int*   epoch  = (const int*)d_in[2];

    const int nRows    = in_sizes[1];               // B = 4,194,304
    const int numTiles = nRows / TILE_ROWS;         // 8192
    const int tailStart = numTiles * TILE_ROWS;     // == nRows here

    float* partial = (float*)d_ws;
    int*   flags   = (int*)((char*)d_ws + GRID1 * sizeof(float));

    int grid = numTiles < GRID1 ? (numTiles > 0 ? numTiles : 1) : GRID1;

    ce_partial_kernel<<<grid, TPB, 0, stream>>>(logits, target, partial, flags, numTiles);
    ce_final_kernel<<<1, TPB, 0, stream>>>(partial, flags, logits, target, epoch,
                                           (float*)d_out, grid, tailStart, nRows);
}